// Layer_1382979469612
// MI455X (gfx1250) — compile-verified
//
#include <hip/hip_runtime.h>
#include <hip/hip_bf16.h>
#include <math.h>

typedef __attribute__((ext_vector_type(2))) float v2f;
typedef __attribute__((ext_vector_type(8))) float v8f;

#define NB    4
#define SIN   1024
#define SOUT  512
#define DIN   512
#define DOUT  256
#define NROWS (NB * SIN)          // 4096 (b,s) rows
#define LDSW  520                 // 512 + pad -> conflict-free A-frag reads

// ---------------------------------------------------------------------------
// Kernel 1: Z = x @ M^T and Rfeat = x @ Wres^T (fused, shared A), + LayerNorm(Z)
// Block: 256 thr (8 waves) handles 16 rows x all 512 output cols (256 Z + 256 R).
// ---------------------------------------------------------------------------
__global__ __launch_bounds__(256) void k_proj_ln(
    const float* __restrict__ x, const float* __restrict__ M,
    const float* __restrict__ Wres, const float* __restrict__ gamma,
    const float* __restrict__ beta,
    float* __restrict__ Zn, float* __restrict__ Rf)
{
    __shared__ float xs[16 * LDSW];   // x slab [16][512]; reused to stage D
    __shared__ float psum[256];
    __shared__ float psum2[256];
    __shared__ float mu_s[16];
    __shared__ float rs_s[16];

    const int tid  = threadIdx.x;
    const int lane = tid & 31;
    const int wave = tid >> 5;
    const int r0   = blockIdx.x * 16;

    // stage x[r0:r0+16, 0:512] into LDS
    for (int e = 0; e < 32; ++e) {
        int idx = tid + e * 256;           // 0..8191
        int row = idx >> 9, col = idx & 511;
        xs[row * LDSW + col] = x[(size_t)(r0 + row) * DIN + col];
    }
    __syncthreads();

    const int lr = lane & 15;              // row within A tile / col within B tile
    const int kh = (lane >> 4) * 2;        // K sub-offset for upper half-wave

    v8f acc[4] = {};                       // 4 output tiles (16x16) per wave
    for (int k0 = 0; k0 < DIN; k0 += 4) {
        v2f a;
        a.x = xs[lr * LDSW + k0 + kh];
        a.y = xs[lr * LDSW + k0 + kh + 1];
#pragma unroll
        for (int q = 0; q < 4; ++q) {
            int ct = wave * 4 + q;                       // 0..31 col tiles
            const float* Wt = (ct < 16) ? M : Wres;      // uniform per tile
            int n0 = (ct & 15) * 16;
            const float* bp = Wt + (size_t)(n0 + lr) * DIN + k0 + kh;
            v2f b; b.x = bp[0]; b.y = bp[1];
            acc[q] = __builtin_amdgcn_wmma_f32_16x16x4_f32(
                false, a, false, b, (short)0, acc[q], false, false);
        }
    }
    __syncthreads();

    // stage D tiles back to LDS as ds[16][512]: cols 0..255 = Z, 256..511 = Rfeat
#pragma unroll
    for (int q = 0; q < 4; ++q) {
        int colb = (wave * 4 + q) * 16 + lr;
#pragma unroll
        for (int r = 0; r < 8; ++r) {
            int row = r + ((lane >> 4) << 3);
            xs[row * LDSW + colb] = acc[q][r];
        }
    }
    __syncthreads();

    // LayerNorm stats over Z (cols 0..255): 16 partials per row
    {
        int row = tid >> 4, seg = tid & 15;
        float s = 0.f, s2 = 0.f;
#pragma unroll
        for (int j = 0; j < 16; ++j) {
            float v = xs[row * LDSW + seg * 16 + j];
            s += v; s2 += v * v;
        }
        psum[tid] = s; psum2[tid] = s2;
    }
    __syncthreads();
    if (tid < 16) {
        float s = 0.f, s2 = 0.f;
#pragma unroll
        for (int j = 0; j < 16; ++j) { s += psum[tid * 16 + j]; s2 += psum2[tid * 16 + j]; }
        float mu  = s * (1.0f / 256.0f);
        float var = s2 * (1.0f / 256.0f) - mu * mu;
        mu_s[tid] = mu;
        rs_s[tid] = rsqrtf(var + 1e-5f);
    }
    __syncthreads();

    // write Zn (normalized) and Rf (raw residual features)
    for (int e = 0; e < 16; ++e) {
        int idx = tid + e * 256;            // 0..4095
        int row = idx >> 8, col = idx & 255;
        float z  = xs[row * LDSW + col];
        float zn = (z - mu_s[row]) * rs_s[row] * gamma[col] + beta[col];
        Zn[(size_t)(r0 + row) * DOUT + col] = zn;
        Rf[(size_t)(r0 + row) * DOUT + col] = xs[row * LDSW + 256 + col];
    }
}

// ---------------------------------------------------------------------------
// Kernel 2: TR[b,s,i] = sum_j Zn[b,s,j] * P[i,j] * cos(2*pi*s/(i*256+j+2)) + Rf[b,s,i]
// Dominant stage (67M rcp+cos, 67M P-element reads). P is tiled through LDS with
// double-buffered GLOBAL_LOAD_ASYNC_TO_LDS_B128 (ASYNCcnt) so the next tile's DMA
// overlaps the current tile's transcendental compute, and all global reads of P
// are fully coalesced 16B chunks instead of lane-scattered 1KB-strided dwords.
// ---------------------------------------------------------------------------
#define JT      64                // j-tile width
#define NTILES  (DOUT / JT)       // 4
#define PSTRIDE 68                // floats per LDS P row: 16B aligned, low-conflict

__device__ __forceinline__ void issue_p_tile(const float* __restrict__ P, int j0,
                                             float* buf, int tid)
{
#pragma unroll
    for (int e = 0; e < 16; ++e) {
        int idx4 = e * 256 + tid;          // 16-byte chunk id, 0..4095
        int r    = idx4 >> 4;              // 16 chunks per 64-col row
        int c4   = (idx4 & 15) * 4;
        const float* g = P + (size_t)r * DOUT + j0 + c4;
        unsigned lds = (unsigned)(unsigned long long)(buf + r * PSTRIDE + c4);
        asm volatile("global_load_async_to_lds_b128 %0, %1, off"
                     :: "v"(lds), "v"(g) : "memory");
    }
}

__global__ __launch_bounds__(256) void k_cosmod(
    const float* __restrict__ Zn, const float* __restrict__ P,
    const float* __restrict__ Rf, float* __restrict__ TR)
{
    __shared__ float zn[NB][DOUT];
    __shared__ __align__(16) float pt[2][DOUT * PSTRIDE];   // double-buffered P tile

    const int s = blockIdx.x;
    const int i = threadIdx.x;

#pragma unroll
    for (int b = 0; b < NB; ++b)
        zn[b][i] = Zn[((size_t)b * SIN + s) * DOUT + i];

    issue_p_tile(P, 0, pt[0], i);          // prefetch tile 0

    const float two_pi_s = 6.283185307179586f * (float)s;
    const float pbase    = (float)(i * DOUT + 2);

    float a0 = 0.f, a1 = 0.f, a2 = 0.f, a3 = 0.f;

    for (int tj = 0; tj < NTILES; ++tj) {
        if (tj + 1 < NTILES) {
            issue_p_tile(P, (tj + 1) * JT, pt[(tj + 1) & 1], i);
            asm volatile("s_wait_asynccnt 0x10" ::: "memory");  // tile tj done
        } else {
            asm volatile("s_wait_asynccnt 0x0" ::: "memory");
        }
        __syncthreads();                   // all waves' DMA for tile tj complete

        const float4* prow4 = (const float4*)(&pt[tj & 1][i * PSTRIDE]);
        float jf = (float)(tj * JT);
#pragma unroll
        for (int q4 = 0; q4 < JT / 4; ++q4) {
            float4 pv = prow4[q4];
            int j = tj * JT + q4 * 4;
            float c0 = __cosf(two_pi_s * __builtin_amdgcn_rcpf(pbase + jf));
            float c1 = __cosf(two_pi_s * __builtin_amdgcn_rcpf(pbase + jf + 1.0f));
            float c2 = __cosf(two_pi_s * __builtin_amdgcn_rcpf(pbase + jf + 2.0f));
            float c3 = __cosf(two_pi_s * __builtin_amdgcn_rcpf(pbase + jf + 3.0f));
            float w0 = pv.x * c0, w1 = pv.y * c1, w2 = pv.z * c2, w3 = pv.w * c3;
            a0 += zn[0][j] * w0 + zn[0][j+1] * w1 + zn[0][j+2] * w2 + zn[0][j+3] * w3;
            a1 += zn[1][j] * w0 + zn[1][j+1] * w1 + zn[1][j+2] * w2 + zn[1][j+3] * w3;
            a2 += zn[2][j] * w0 + zn[2][j+1] * w1 + zn[2][j+2] * w2 + zn[2][j+3] * w3;
            a3 += zn[3][j] * w0 + zn[3][j+1] * w1 + zn[3][j+2] * w2 + zn[3][j+3] * w3;
            jf += 4.0f;
        }
        __syncthreads();                   // buffer may be overwritten next iter
    }

    TR[((size_t)0 * SIN + s) * DOUT + i] = a0 + Rf[((size_t)0 * SIN + s) * DOUT + i];
    TR[((size_t)1 * SIN + s) * DOUT + i] = a1 + Rf[((size_t)1 * SIN + s) * DOUT + i];
    TR[((size_t)2 * SIN + s) * DOUT + i] = a2 + Rf[((size_t)2 * SIN + s) * DOUT + i];
    TR[((size_t)3 * SIN + s) * DOUT + i] = a3 + Rf[((size_t)3 * SIN + s) * DOUT + i];
}

// ---------------------------------------------------------------------------
// Kernel 3: Out[b,t,d] = sum_s TR[b,s,d] * Linker[s,t]   (residual already folded)
// Block = (b, 16-t slab): A = Linker^T slab staged in LDS (64KB), WMMA over K=1024.
// ---------------------------------------------------------------------------
__global__ __launch_bounds__(256) void k_linker(
    const float* __restrict__ TR, const float* __restrict__ Linker,
    float* __restrict__ Out)
{
    __shared__ float lk[SIN * 16];   // Linker[:, t0:t0+16], row-major [k][16]

    const int tid   = threadIdx.x;
    const int lane  = tid & 31;
    const int wave  = tid >> 5;
    const int bb    = blockIdx.x >> 5;       // batch 0..3
    const int tslab = blockIdx.x & 31;       // 0..31
    const int t0    = tslab * 16;

    for (int e = 0; e < 64; ++e) {
        int idx = tid + e * 256;             // 0..16383
        int k = idx >> 4, c = idx & 15;
        lk[idx] = Linker[(size_t)k * SOUT + t0 + c];
    }
    __syncthreads();

    const int lr = lane & 15;
    const int kh = (lane >> 4) * 2;
    const float* trb = TR + (size_t)bb * SIN * DOUT;

    v8f acc[2] = {};
    for (int k0 = 0; k0 < SIN; k0 += 4) {
        int kk = k0 + kh;
        v2f a;
        a.x = lk[kk * 16 + lr];              // A[t][k] = Linker[k][t]
        a.y = lk[(kk + 1) * 16 + lr];
#pragma unroll
        for (int q = 0; q < 2; ++q) {
            int d0 = (wave * 2 + q) * 16;
            v2f b;
            b.x = trb[(size_t)kk * DOUT + d0 + lr];
            b.y = trb[(size_t)(kk + 1) * DOUT + d0 + lr];
            acc[q] = __builtin_amdgcn_wmma_f32_16x16x4_f32(
                false, a, false, b, (short)0, acc[q], false, false);
        }
    }

#pragma unroll
    for (int q = 0; q < 2; ++q) {
        int d0 = (wave * 2 + q) * 16;
#pragma unroll
        for (int r = 0; r < 8; ++r) {
            int row = t0 + r + ((lane >> 4) << 3);
            Out[(size_t)bb * SOUT * DOUT + (size_t)row * DOUT + d0 + lr] = acc[q][r];
        }
    }
}

// ---------------------------------------------------------------------------
extern "C" void kernel_launch(void* const* d_in, const int* in_sizes, int n_in,
                              void* d_out, int out_size, void* d_ws, size_t ws_size,
                              hipStream_t stream) {
    const float* x      = (const float*)d_in[0];
    const float* M      = (const float*)d_in[1];
    const float* P      = (const float*)d_in[2];
    const float* Linker = (const float*)d_in[3];
    const float* gamma  = (const float*)d_in[4];
    const float* beta   = (const float*)d_in[5];
    const float* Wres   = (const float*)d_in[6];
    float* out = (float*)d_out;

    float* Zn = (float*)d_ws;                      // [4096, 256]
    float* Rf = Zn + (size_t)NROWS * DOUT;         // [4096, 256]
    float* TR = Rf + (size_t)NROWS * DOUT;         // [4096, 256]   total 12 MB

    k_proj_ln<<<NROWS / 16, 256, 0, stream>>>(x, M, Wres, gamma, beta, Zn, Rf);
    k_cosmod <<<SIN, 256, 0, stream>>>(Zn, P, Rf, TR);
    k_linker <<<NB * (SOUT / 16), 256, 0, stream>>>(TR, Linker, out);
}